// KeypointPostProcessor_37306085933630
// MI455X (gfx1250) — compile-verified
//
#include <hip/hip_runtime.h>

// Keypoint post-processing: B=256, Q=2048, K=17 (51 channels/query).
// Memory-bound: ~223 MB traffic, HBM floor ~9.6 us @ 23.3 TB/s. No matrix op
// -> no WMMA. This version makes the channel decode loop-invariant per thread
// (c = tid % 51 fixed) so steady state is ~7 VALU + 3 ds per 8 bytes moved,
// keeping us under the VALU roofline while streaming coalesced NT b32.

#define B_    256
#define Q_    2048
#define NC    51u
#define NXY   34u
#define QB    60          // queries per full block (5 queries x 12 passes)
#define NP    12          // passes per full block
#define NTH   256         // 8 wave32; lanes 0..254 active in stream
#define QTAIL 8           // 2048 - 34*60
#define NFULL 34          // full blocks per batch image

typedef float v4f __attribute__((ext_vector_type(4)));

__global__ __launch_bounds__(NTH) void kp_post_kernel(
    const float* __restrict__ kp,            // [B,Q,51]
    const float* __restrict__ boxes,         // [B,Q,4]
    const unsigned char* __restrict__ pmask, // [B,Q] (jax bool = 1 byte)
    const int* __restrict__ orig,            // [B,2] (h,w)
    float* __restrict__ out)                 // [B,Q,51]
{
    // Per-query record, 8 floats (32B): {x1*mk, y1*mk, bw*mk, bh*mk, mk, mk, unset, unset}
    __shared__ float spar[QB * 8];

    const int b   = blockIdx.y;
    const int bx  = blockIdx.x;
    const int q0  = bx * QB;                 // tail block: q0 = 2040
    const int tid = threadIdx.x;

    // Per-batch scalars: uniform -> SGPRs.
    const int h_i   = orig[2 * b + 0];
    const int w_i   = orig[2 * b + 1];
    const int max_i = (h_i > w_i) ? h_i : w_i;
    const float left_pad = (float)((max_i - w_i) >> 1);
    const float top_pad  = (float)((max_i - h_i) >> 1);
    const float max_size = (float)max_i;
    const float img_w    = (float)w_i;
    const float img_h    = (float)h_i;

    // ---- Stage 1: one lane per query -> packed params in LDS ----
    const int nq = (bx < NFULL) ? QB : QTAIL;
    if (tid < nq) {
        const int q = q0 + tid;              // always < Q_ by construction
        const v4f bxv = ((const v4f*)boxes)[(size_t)b * Q_ + q];   // 16B aligned
        const float cx = bxv.x, cy = bxv.y, w = bxv.z, h = bxv.w;
        const float x1 = fminf(fmaxf(fmaf(cx - 0.5f * w, max_size, -left_pad), 0.0f), img_w);
        const float y1 = fminf(fmaxf(fmaf(cy - 0.5f * h, max_size, -top_pad ), 0.0f), img_h);
        const float x2 = fminf(fmaxf(fmaf(cx + 0.5f * w, max_size, -left_pad), 0.0f), img_w);
        const float y2 = fminf(fmaxf(fmaf(cy + 0.5f * h, max_size, -top_pad ), 0.0f), img_h);
        const float mk = pmask[(size_t)b * Q_ + q] ? 0.0f : 1.0f;
        float* rec = &spar[tid * 8];
        rec[0] = x1 * mk;
        rec[1] = y1 * mk;
        rec[2] = (x2 - x1) * mk;
        rec[3] = (y2 - y1) * mk;
        rec[4] = mk;
        rec[5] = mk;                          // duplicate: mk readable at base+4 for either parity
    }
    __syncthreads();

    const size_t baseF = ((size_t)b * Q_ + q0) * NC;

    if (bx < NFULL) {
        // ---- Main path: no bounds checks, loop-invariant channel decode ----
        if (tid < 255) {
            const unsigned c    = (unsigned)tid % NC;   // one magic div, once
            const unsigned qofs = (unsigned)tid / NC;   // 0..4
            const unsigned lo   = c & 1u;               // 0=x, 1=y
            const bool isCoord  = c < NXY;
            const float lim     = lo ? img_h : img_w;

            const float* a = &spar[qofs * 8u + lo];     // t=a[0], s=a[2], mk=a[4]
            const float* __restrict__ gin  = kp  + baseF + tid;
            float*       __restrict__ gout = out + baseF + tid;

            // Issue all 12 NT loads up front: deep MLP, imm offsets p*255 floats.
            float e[NP];
#pragma unroll
            for (int p = 0; p < NP; ++p)
                e[p] = __builtin_nontemporal_load(gin + p * 255);

#pragma unroll
            for (int p = 0; p < NP; ++p) {
                const float* ap = a + p * 40;           // 5 records per pass (DS imm offsets)
                const float tt = ap[0];
                const float ss = ap[2];
                const float mk = ap[4];
                const float coord = fminf(fmaxf(fmaf(e[p], ss, tt), 0.0f), lim);
                const float r = isCoord ? coord : e[p] * mk;
                __builtin_nontemporal_store(r, gout + p * 255);
            }
        }
    } else {
        // ---- Tail block (8 queries, 408 floats): rare generic path ----
        for (int i = tid; i < QTAIL * (int)NC; i += NTH) {
            const unsigned ql = (unsigned)i / NC;
            const unsigned cc = (unsigned)i - ql * NC;
            const unsigned lo = cc & 1u;
            const float* ap = &spar[ql * 8u + lo];
            const float e  = __builtin_nontemporal_load(kp + baseF + i);
            const float tt = ap[0];
            const float ss = ap[2];
            const float mk = ap[4];
            const float lim = lo ? img_h : img_w;
            const float coord = fminf(fmaxf(fmaf(e, ss, tt), 0.0f), lim);
            const float r = (cc < NXY) ? coord : e * mk;
            __builtin_nontemporal_store(r, out + baseF + i);
        }
    }
}

extern "C" void kernel_launch(void* const* d_in, const int* in_sizes, int n_in,
                              void* d_out, int out_size, void* d_ws, size_t ws_size,
                              hipStream_t stream) {
    (void)in_sizes; (void)n_in; (void)out_size; (void)d_ws; (void)ws_size;
    const float*         kp    = (const float*)d_in[0];
    const float*         boxes = (const float*)d_in[1];
    const unsigned char* mask  = (const unsigned char*)d_in[2];
    const int*           orig  = (const int*)d_in[3];
    float*               outp  = (float*)d_out;

    dim3 grid(NFULL + 1, B_, 1);   // 35 x 256 blocks
    dim3 block(NTH, 1, 1);
    kp_post_kernel<<<grid, block, 0, stream>>>(kp, boxes, mask, orig, outp);
}